// YOLOLayer_18975165514227
// MI455X (gfx1250) — compile-verified
//
#include <hip/hip_runtime.h>

// YOLO head: (bs, na*nc, fy, fx) -> (bs, na*fy*fx, nc) transpose fused with
// sigmoid/exp transforms. Memory-bound: ~251 MB traffic -> ~10.8 us floor at
// 23.3 TB/s. Strategy: per-block LDS tile transpose using CDNA5 async
// global->LDS loads (ASYNCcnt), coalesced on both global sides; vectorized
// b128 LDS reads + global stores on the output side.

#define NA 4
#define NC 85
#define FX 76
#define FY 76
#define SP (FX * FY)            // 5776 spatial positions
#define TILE_ELEMS (NC * FX)    // 6460 floats = 25840 B LDS
#define TILE_VECS (TILE_ELEMS / 4)  // 1615 float4s (6460 % 4 == 0)
#define BLOCK 256

#if __has_builtin(__builtin_amdgcn_global_load_async_to_lds_b32) && \
    __has_builtin(__builtin_amdgcn_s_wait_asynccnt)
#define USE_ASYNC_LDS 1
#else
#define USE_ASYNC_LDS 0
#endif

typedef __attribute__((address_space(1))) int global_i32;
typedef __attribute__((address_space(3))) int lds_i32;

__device__ __forceinline__ float fast_sigmoid(float v) {
    // v_exp_f32 + v_rcp_f32
    return __builtin_amdgcn_rcpf(1.0f + __expf(-v));
}

__device__ __forceinline__ float transform(float v, int c, int s,
                                           float aw, float ah, float tyf) {
    if (c >= 4) return fast_sigmoid(v);                      // 81/85 of elems
    if (c == 0) return (fast_sigmoid(v) + (float)s) * 8.0f;  // px
    if (c == 1) return (fast_sigmoid(v) + tyf) * 8.0f;       // py
    if (c == 2) return __expf(v) * aw;                       // pw
    return __expf(v) * ah;                                   // ph
}

__global__ __launch_bounds__(BLOCK) void yolo_head_kernel(
    const float* __restrict__ x,        // (16, 340, 76, 76)
    const float* __restrict__ anchors,  // (4, 2), already divided by STRIDE
    float* __restrict__ out)            // (16, 4*76*76, 85)
{
    __shared__ float tile[TILE_ELEMS];  // spatial-major: tile[s*NC + c]

    const int tid = (int)threadIdx.x;
    const int bid = (int)blockIdx.x;
    const int ty  = bid % FY;           // spatial row (y)
    const int a   = (bid / FY) % NA;    // anchor
    const int b   = bid / (FY * NA);    // batch

    // input base: x[b][a*NC + c][ty][0..75], channel stride = SP floats
    const float* inp = x + ((size_t)(b * (NA * NC) + a * NC)) * SP + (size_t)ty * FX;

    // ---- Load phase: channel-major global -> spatial-major LDS (transpose).
    // i = c*FX + s; consecutive lanes have consecutive s -> coalesced 304B
    // runs per channel, combined into full lines by hardware. LDS deposit at
    // s*85+c: bank stride 21 (coprime 64) -> conflict-free.
#pragma unroll 4
    for (int i = tid; i < TILE_ELEMS; i += BLOCK) {
        const int c = i / FX;
        const int s = i - c * FX;
#if USE_ASYNC_LDS
        __builtin_amdgcn_global_load_async_to_lds_b32(
            (global_i32*)(inp + (size_t)c * SP + s),
            (lds_i32*)(&tile[s * NC + c]),
            /*offset=*/0, /*cpol=*/0);
#else
        tile[s * NC + c] = inp[(size_t)c * SP + s];
#endif
    }
#if USE_ASYNC_LDS
    __builtin_amdgcn_s_wait_asynccnt(0);  // own wave's async LDS writes done
#endif
    __syncthreads();                      // cross-wave visibility

    // Reference math with fx/fy cancellation:
    //  px = (sigmoid(v0) + x) * 8 ; py = (sigmoid(v1) + y) * 8
    //  pw = exp(v2) * anchors[a][0] * 8 ; ph = exp(v3) * anchors[a][1] * 8
    //  rest = sigmoid(v)
    const float aw  = anchors[a * 2 + 0] * 8.0f;
    const float ah  = anchors[a * 2 + 1] * 8.0f;
    const float tyf = (float)ty;

    // output base: out flat index = ((b*NA+a)*SP + ty*FX + s)*NC + c.
    // Block slab = 6460 floats, 16B aligned (6460*4 % 16 == 0).
    float* o = out + ((size_t)(b * NA + a) * SP + (size_t)ty * FX) * NC;

    // ---- Store phase: vectorized. Each thread handles 4 consecutive output
    // floats: ds_load_b128 (stride-1, conflict-free) + global_store_b128
    // (512 B/wave contiguous).
#pragma unroll 2
    for (int iv = tid; iv < TILE_VECS; iv += BLOCK) {
        const int i0 = iv * 4;
        const float4 v4 = *reinterpret_cast<const float4*>(&tile[i0]);

        int s = i0 / NC;
        int c = i0 - s * NC;
        float vv[4] = {v4.x, v4.y, v4.z, v4.w};
        float rr[4];
#pragma unroll
        for (int j = 0; j < 4; ++j) {
            rr[j] = transform(vv[j], c, s, aw, ah, tyf);
            if (++c == NC) { c = 0; ++s; }   // channel wrap inside the quad
        }
        float4 r4;
        r4.x = rr[0]; r4.y = rr[1]; r4.z = rr[2]; r4.w = rr[3];
        *reinterpret_cast<float4*>(&o[i0]) = r4;
    }
}

extern "C" void kernel_launch(void* const* d_in, const int* in_sizes, int n_in,
                              void* d_out, int out_size, void* d_ws, size_t ws_size,
                              hipStream_t stream) {
    const float* x       = (const float*)d_in[0];
    const float* anchors = (const float*)d_in[1];
    float* out           = (float*)d_out;

    const int bs = in_sizes[0] / (NA * NC * SP);  // 16
    dim3 grid(bs * NA * FY);                      // 4864 blocks
    yolo_head_kernel<<<grid, BLOCK, 0, stream>>>(x, anchors, out);
}